// TopKGate_40707700032214
// MI455X (gfx1250) — compile-verified
//
#include <hip/hip_runtime.h>
#include <hip/hip_bf16.h>

typedef __attribute__((ext_vector_type(2))) float v2f;
typedef __attribute__((ext_vector_type(8))) float v8f;

#define D_DIM 4096
#define E_DIM 64
#define ROWS_PER_BLOCK 64   // 4 waves x 16 rows
#define KT 64               // K tile staged in LDS per iteration
#define LSTR (KT + 4)       // LDS dword stride: 16B-aligned float4 stores, spreads banks
#define LGSTR (E_DIM + 1)   // logits scratch stride: conflict-free row scans

__global__ __launch_bounds__(128) void topk_gate_kernel(
    const float* __restrict__ x, const float* __restrict__ W,
    float* __restrict__ outW, int* __restrict__ outIdx)
{
    __shared__ float xs[ROWS_PER_BLOCK * LSTR];   // x tile   64 x 64
    __shared__ float ws[E_DIM * LSTR];            // W tile   64 x 64
    __shared__ float lg[ROWS_PER_BLOCK * LGSTR];  // logits   64 x 64 (+pad)

    const int tid   = threadIdx.x;
    const int wave  = tid >> 5;
    const int lane  = tid & 31;
    const int lm    = lane & 15;      // M (A) / N (B) index within 16
    const int lh    = lane >> 4;      // lane-half selects K pair / row half
    const int klane = lh * 2;         // K base for this lane's fragment
    const int rowBase = blockIdx.x * ROWS_PER_BLOCK;
    const int wrow  = wave * 16 + lm; // this lane's A-matrix row in the block tile

    v8f acc0 = {}, acc1 = {}, acc2 = {}, acc3 = {};

    for (int k0 = 0; k0 < D_DIM; k0 += KT) {
        __syncthreads();  // previous tile's fragments consumed before overwrite
        // Cooperative staging: 64x64 f32 x-tile and W-tile, coalesced b128 loads.
        // 1024 float4 slots each, 128 threads -> 8 slots/thread.
#pragma unroll
        for (int i = 0; i < 8; ++i) {
            int f  = tid + i * 128;
            int r  = f >> 4;           // row 0..63
            int c4 = (f & 15) << 2;    // col 0..60 step 4
            float4 xv = *(const float4*)(x + (size_t)(rowBase + r) * D_DIM + k0 + c4);
            float4 wv = *(const float4*)(W + (size_t)r * D_DIM + k0 + c4);
            *(float4*)&xs[r * LSTR + c4] = xv;
            *(float4*)&ws[r * LSTR + c4] = wv;
        }
        __syncthreads();

#pragma unroll
        for (int kk = 0; kk < KT; kk += 4) {
            // fp32 WMMA fragment layout (ISA 7.12.2): lane L holds index L%16,
            // VGPR v holds K = (L/16)*2 + v  -> one aligned 8-byte LDS read each.
            v2f a  = *(const v2f*)&xs[wrow * LSTR + kk + klane];
            v2f b0 = *(const v2f*)&ws[( 0 + lm) * LSTR + kk + klane];
            v2f b1 = *(const v2f*)&ws[(16 + lm) * LSTR + kk + klane];
            v2f b2 = *(const v2f*)&ws[(32 + lm) * LSTR + kk + klane];
            v2f b3 = *(const v2f*)&ws[(48 + lm) * LSTR + kk + klane];
            acc0 = __builtin_amdgcn_wmma_f32_16x16x4_f32(false, a, false, b0, (short)0, acc0, false, false);
            acc1 = __builtin_amdgcn_wmma_f32_16x16x4_f32(false, a, false, b1, (short)0, acc1, false, false);
            acc2 = __builtin_amdgcn_wmma_f32_16x16x4_f32(false, a, false, b2, (short)0, acc2, false, false);
            acc3 = __builtin_amdgcn_wmma_f32_16x16x4_f32(false, a, false, b3, (short)0, acc3, false, false);
        }
    }

    // Spill logits: C-tile layout -> row = wave*16 + 8*lh + j, col = 16*t + lm
    __syncthreads();
#pragma unroll
    for (int j = 0; j < 8; ++j) {
        int r = wave * 16 + lh * 8 + j;
        lg[r * LGSTR +  0 + lm] = acc0[j];
        lg[r * LGSTR + 16 + lm] = acc1[j];
        lg[r * LGSTR + 32 + lm] = acc2[j];
        lg[r * LGSTR + 48 + lm] = acc3[j];
    }
    __syncthreads();

    // One thread per row: top-2 scan (strict > == lowest-index tie-break, as
    // in jax.lax.top_k), softmax over the 2 selected, scatter + indices.
    if (tid < ROWS_PER_BLOCK) {
        const float* row = &lg[tid * LGSTR];
        float v0 = -__builtin_inff(), v1 = -__builtin_inff();
        int   i0 = 0, i1 = 0;
        for (int e = 0; e < E_DIM; ++e) {
            float v = row[e];
            if (v > v0)      { v1 = v0; i1 = i0; v0 = v; i0 = e; }
            else if (v > v1) { v1 = v;  i1 = e; }
        }
        float t  = __expf(v1 - v0);        // <= 1, numerically safe
        float w0 = 1.0f / (1.0f + t);
        float w1 = 1.0f - w0;

        int    grow = rowBase + tid;
        float* orow = outW + (size_t)grow * E_DIM;
#pragma unroll
        for (int c4 = 0; c4 < E_DIM; c4 += 4) {
            float4 z = make_float4(0.f, 0.f, 0.f, 0.f);
            if ((unsigned)(i0 - c4) < 4u) (&z.x)[i0 - c4] = w0;
            if ((unsigned)(i1 - c4) < 4u) (&z.x)[i1 - c4] = w1;
            *(float4*)(orow + c4) = z;
        }
        outIdx[grow * 2 + 0] = i0;
        outIdx[grow * 2 + 1] = i1;
    }
}

extern "C" void kernel_launch(void* const* d_in, const int* in_sizes, int n_in,
                              void* d_out, int out_size, void* d_ws, size_t ws_size,
                              hipStream_t stream) {
    const float* x = (const float*)d_in[0];   // [N, 4096] fp32
    const float* W = (const float*)d_in[1];   // [64, 4096] fp32
    const int N = in_sizes[0] / D_DIM;        // 16384

    float* outW  = (float*)d_out;                          // [N, 64] full_weights
    int*   outIdx = (int*)((float*)d_out + (size_t)N * E_DIM); // [N, 2] int32 indices

    dim3 grid(N / ROWS_PER_BLOCK);
    dim3 block(128);
    hipLaunchKernelGGL(topk_gate_kernel, grid, block, 0, stream, x, W, outW, outIdx);
}